// GraphNet_65738769433238
// MI455X (gfx1250) — compile-verified
//
#include <hip/hip_runtime.h>
#include <math.h>

#define Hd 128
#define Gd 64
#define Cd 10
#define NEG_SLOPE 0.2f

typedef float v2f __attribute__((ext_vector_type(2)));
typedef float v8f __attribute__((ext_vector_type(8)));

// Monotone float atomic max via integer atomics (lowers to global_atomic_max_i32 /
// global_atomic_min_u32, no CAS loop). Valid with -inf init.
__device__ __forceinline__ void atomicMaxF(float* addr, float v) {
    if (v >= 0.0f) atomicMax((int*)addr, __float_as_int(v));
    else           atomicMin((unsigned int*)addr, __float_as_uint(v));
}

// ---------------- init kernels ----------------
__global__ void init0_kernel(float* z, float* counts) {
    int idx = blockIdx.x * blockDim.x + threadIdx.x;
    if (idx < Gd * 2 * Hd) z[idx] = 0.0f;
    if (idx < Gd)          counts[idx] = 0.0f;
}

__global__ void counts_kernel(const int* __restrict__ batch, float* counts, int nN) {
    int i = blockIdx.x * blockDim.x + threadIdx.x;
    if (i < nN) atomicAdd(&counts[batch[i]], 1.0f);
}

// zero scatter-output buffer, init segment max/sum and pool accumulators
__global__ void init_layer_kernel(float* outbuf, float* segmax, float* segsum,
                                  float* pmax, float* psum, int nN) {
    int idx = blockIdx.x * blockDim.x + threadIdx.x;
    if (idx < nN * Hd) outbuf[idx] = 0.0f;
    if (idx < nN) { segmax[idx] = -INFINITY; segsum[idx] = 0.0f; }
    if (idx < Gd * Hd) { pmax[idx] = -INFINITY; psum[idx] = 0.0f; }
}

// ---------------- layer-1 transform: h = concat(x,pos) @ W1 ----------------
__global__ void l1_transform_kernel(const float* __restrict__ x, const float* __restrict__ pos,
                                    const float* __restrict__ W1, float* __restrict__ h, int nN) {
    int idx = blockIdx.x * blockDim.x + threadIdx.x;
    if (idx >= nN * Hd) return;
    int i = idx / Hd, j = idx - i * Hd;
    float acc = 0.0f;
#pragma unroll
    for (int k = 0; k < 3; ++k) acc = fmaf(x[i * 3 + k], W1[k * Hd + j], acc);
#pragma unroll
    for (int k = 0; k < 3; ++k) acc = fmaf(pos[i * 3 + k], W1[(3 + k) * Hd + j], acc);
    h[idx] = acc;
}

// ---------------- WMMA fp32 GEMM: C[M,128] = A[M,128] @ B[128,128] ----------------
// One wave computes one 16x16 tile; K swept 4 at a time with v_wmma_f32_16x16x4_f32.
__global__ void gemm_wmma_kernel(const float* __restrict__ A, const float* __restrict__ B,
                                 float* __restrict__ C, int M) {
    int wid  = (blockIdx.x * blockDim.x + threadIdx.x) >> 5;
    int lane = threadIdx.x & 31;
    const int ntn = Hd / 16;            // 8 col tiles
    int row_tile = wid / ntn;
    int col_tile = wid - row_tile * ntn;
    if (row_tile * 16 >= M) return;     // wave-uniform: EXEC stays all-ones
    int m0 = row_tile * 16, n0 = col_tile * 16;
    int half = lane >> 4, lid = lane & 15;

    v8f acc = {};
#pragma unroll
    for (int k = 0; k < Hd; k += 4) {
        // A 16x4 tile: lanes 0-15 -> K={k,k+1}, lanes 16-31 -> K={k+2,k+3}
        const float* ap = A + (size_t)(m0 + lid) * Hd + k + 2 * half;
        v2f a; a.x = ap[0]; a.y = ap[1];
        // B 4x16 tile: same K split, N striped across lanes
        const float* bp = B + (size_t)(k + 2 * half) * Hd + n0 + lid;
        v2f b; b.x = bp[0]; b.y = bp[Hd];
        acc = __builtin_amdgcn_wmma_f32_16x16x4_f32(
            /*neg_a=*/false, a, /*neg_b=*/false, b,
            /*c_mod=*/(short)0, acc, /*reuse_a=*/false, /*reuse_b=*/false);
    }
    // C/D layout: VGPR r -> M = r (lanes 0-15) / r+8 (lanes 16-31), N = lid
#pragma unroll
    for (int r = 0; r < 8; ++r)
        C[(size_t)(m0 + r + 8 * half) * Hd + n0 + lid] = acc[r];
}

// ---------------- attention logits: ls[i]=h[i]·a_src, ld[i]=h[i]·a_dst ----------------
__global__ void logits_kernel(const float* __restrict__ h, const float* __restrict__ a_src,
                              const float* __restrict__ a_dst, float* ls, float* ld, int nN) {
    int wave = (blockIdx.x * blockDim.x + threadIdx.x) >> 5;
    int lane = threadIdx.x & 31;
    if (wave >= nN) return;
    const float* hr = h + (size_t)wave * Hd;
    float s = 0.0f, d = 0.0f;
#pragma unroll
    for (int k = lane; k < Hd; k += 32) {
        float v = hr[k];
        s = fmaf(v, a_src[k], s);
        d = fmaf(v, a_dst[k], d);
    }
#pragma unroll
    for (int off = 16; off > 0; off >>= 1) {
        s += __shfl_down(s, off, 32);
        d += __shfl_down(d, off, 32);
    }
    if (lane == 0) { ls[wave] = s; ld[wave] = d; }
}

// ---------------- per-edge segment max of leaky-relu logits ----------------
__global__ void edge_max_kernel(const int* __restrict__ src, const int* __restrict__ dst,
                                const float* __restrict__ ls, const float* __restrict__ ld,
                                float* segmax, int nE, int nN) {
    int e = blockIdx.x * blockDim.x + threadIdx.x;
    if (e >= nE + nN) return;
    int s, d;
    if (e < nE) { s = src[e]; d = dst[e]; } else { s = d = e - nE; }
    float v = ls[s] + ld[d];
    v = (v > 0.0f) ? v : NEG_SLOPE * v;
    atomicMaxF(&segmax[d], v);
}

// ---------------- per-edge exp + segment sum (stores numerator) ----------------
__global__ void edge_exp_kernel(const int* __restrict__ src, const int* __restrict__ dst,
                                const float* __restrict__ ls, const float* __restrict__ ld,
                                const float* __restrict__ segmax, float* segsum,
                                float* __restrict__ ex, int nE, int nN) {
    int e = blockIdx.x * blockDim.x + threadIdx.x;
    if (e >= nE + nN) return;
    int s, d;
    if (e < nE) { s = src[e]; d = dst[e]; } else { s = d = e - nE; }
    float v = ls[s] + ld[d];
    v = (v > 0.0f) ? v : NEG_SLOPE * v;
    float t = __expf(v - segmax[d]);
    ex[e] = t;
    atomicAdd(&segsum[d], t);
}

// ---------------- weighted scatter: out[dst] += alpha * h[src] ----------------
// One wave per edge, float4 per lane (32*4 = 128 channels), coalesced 512B gather.
__global__ void scatter_kernel(const int* __restrict__ src, const int* __restrict__ dst,
                               const float* __restrict__ ex, const float* __restrict__ segsum,
                               const float* __restrict__ h, float* out, int nE, int nN) {
    int gid  = blockIdx.x * blockDim.x + threadIdx.x;
    int e    = gid >> 5;
    int lane = gid & 31;
    if (e >= nE + nN) return;
    int s, d;
    if (e < nE) { s = src[e]; d = dst[e]; } else { s = d = e - nE; }
    float alpha = ex[e] / segsum[d];
    float4 v = ((const float4*)(h + (size_t)s * Hd))[lane];
    float* op = out + (size_t)d * Hd + lane * 4;
    atomicAdd(op + 0, alpha * v.x);
    atomicAdd(op + 1, alpha * v.y);
    atomicAdd(op + 2, alpha * v.z);
    atomicAdd(op + 3, alpha * v.w);
}

// ---------------- bias + relu in place, fused graph pooling ----------------
__global__ void finalize_pool_kernel(float* __restrict__ out, const float* __restrict__ b,
                                     const int* __restrict__ batch,
                                     float* pmax, float* psum, int nN) {
    int idx = blockIdx.x * blockDim.x + threadIdx.x;
    if (idx >= nN * Hd) return;
    int i = idx / Hd, c = idx - i * Hd;
    float v = out[idx] + b[c];
    v = fmaxf(v, 0.0f);
    out[idx] = v;
    int g = batch[i];
    atomicMaxF(&pmax[g * Hd + c], v);
    atomicAdd(&psum[g * Hd + c], v);
}

// ---------------- z += concat(gmp, gap) ----------------
__global__ void accum_z_kernel(const float* __restrict__ pmax, const float* __restrict__ psum,
                               const float* __restrict__ counts, float* z) {
    int idx = blockIdx.x * blockDim.x + threadIdx.x;
    if (idx >= Gd * 2 * Hd) return;
    int g = idx / (2 * Hd), c = idx - g * 2 * Hd;
    float v = (c < Hd) ? pmax[g * Hd + c] : psum[g * Hd + (c - Hd)] / counts[g];
    z[idx] += v;
}

// ---------------- tiny dense head ----------------
__global__ void fc_kernel(const float* __restrict__ in, const float* __restrict__ W,
                          const float* __restrict__ bias, float* __restrict__ out,
                          int K, int J, int relu) {
    int idx = blockIdx.x * blockDim.x + threadIdx.x;
    if (idx >= Gd * J) return;
    int g = idx / J, j = idx - g * J;
    float acc = bias[j];
    for (int k = 0; k < K; ++k) acc = fmaf(in[g * K + k], W[k * J + j], acc);
    if (relu) acc = fmaxf(acc, 0.0f);
    out[idx] = acc;
}

__global__ void logsoftmax_kernel(const float* __restrict__ in, float* __restrict__ out) {
    int g = blockIdx.x * blockDim.x + threadIdx.x;
    if (g >= Gd) return;
    float m = -INFINITY;
#pragma unroll
    for (int c = 0; c < Cd; ++c) m = fmaxf(m, in[g * Cd + c]);
    float s = 0.0f;
#pragma unroll
    for (int c = 0; c < Cd; ++c) s += __expf(in[g * Cd + c] - m);
    float lse = m + __logf(s);
#pragma unroll
    for (int c = 0; c < Cd; ++c) out[g * Cd + c] = in[g * Cd + c] - lse;
}

extern "C" void kernel_launch(void* const* d_in, const int* in_sizes, int n_in,
                              void* d_out, int out_size, void* d_ws, size_t ws_size,
                              hipStream_t stream) {
    const float* x    = (const float*)d_in[0];
    const float* pos  = (const float*)d_in[1];
    const int*   eidx = (const int*)d_in[2];
    const int*   batch= (const int*)d_in[3];

    const float* W[3]    = {(const float*)d_in[4],  (const float*)d_in[8],  (const float*)d_in[12]};
    const float* asrc[3] = {(const float*)d_in[5],  (const float*)d_in[9],  (const float*)d_in[13]};
    const float* adst[3] = {(const float*)d_in[6],  (const float*)d_in[10], (const float*)d_in[14]};
    const float* bias[3] = {(const float*)d_in[7],  (const float*)d_in[11], (const float*)d_in[15]};
    const float* lw1 = (const float*)d_in[16]; const float* lb1 = (const float*)d_in[17];
    const float* lw2 = (const float*)d_in[18]; const float* lb2 = (const float*)d_in[19];
    const float* lw3 = (const float*)d_in[20]; const float* lb3 = (const float*)d_in[21];

    const int nN = in_sizes[0] / 3;       // 50000
    const int nE = in_sizes[2] / 2;       // 1600000
    const int EP = nE + nN;               // edges + self loops
    const int* srcp = eidx;
    const int* dstp = eidx + nE;

    // workspace carve-out (floats)
    float* ws = (float*)d_ws;
    size_t off = 0;
    float* hA     = ws + off; off += (size_t)nN * Hd;
    float* hB     = ws + off; off += (size_t)nN * Hd;
    float* ls     = ws + off; off += nN;
    float* ld     = ws + off; off += nN;
    float* segmax = ws + off; off += nN;
    float* segsum = ws + off; off += nN;
    float* ex     = ws + off; off += EP;
    float* pmax   = ws + off; off += Gd * Hd;
    float* psum   = ws + off; off += Gd * Hd;
    float* counts = ws + off; off += Gd;
    float* z      = ws + off; off += Gd * 2 * Hd;
    float* t1     = ws + off; off += Gd * Hd;
    float* t2     = ws + off; off += Gd * (Hd / 2);
    float* t3     = ws + off; off += Gd * Cd;

    const int BLK = 256;
    int gNH   = (nN * Hd + BLK - 1) / BLK;              // 25000
    int gN    = (nN + BLK - 1) / BLK;
    int gNwav = (nN * 32 + BLK - 1) / BLK;              // wave-per-node
    int gEP   = (EP + BLK - 1) / BLK;
    int gEPw  = ((size_t)EP * 32 + BLK - 1) / BLK;      // wave-per-edge
    int gZ    = (Gd * 2 * Hd + BLK - 1) / BLK;
    int gGemm = ((nN / 16) * (Hd / 16) * 32 + BLK - 1) / BLK; // 1 wave / 16x16 tile

    init0_kernel<<<gZ, BLK, 0, stream>>>(z, counts);
    counts_kernel<<<gN, BLK, 0, stream>>>(batch, counts, nN);

    for (int layer = 0; layer < 3; ++layer) {
        // transform: hA = prev_act @ W   (prev_act is x|pos for layer 0, hB otherwise)
        if (layer == 0)
            l1_transform_kernel<<<gNH, BLK, 0, stream>>>(x, pos, W[0], hA, nN);
        else
            gemm_wmma_kernel<<<gGemm, BLK, 0, stream>>>(hB, W[layer], hA, nN);

        // attention + scatter: hA -> hB
        init_layer_kernel<<<gNH, BLK, 0, stream>>>(hB, segmax, segsum, pmax, psum, nN);
        logits_kernel<<<gNwav, BLK, 0, stream>>>(hA, asrc[layer], adst[layer], ls, ld, nN);
        edge_max_kernel<<<gEP, BLK, 0, stream>>>(srcp, dstp, ls, ld, segmax, nE, nN);
        edge_exp_kernel<<<gEP, BLK, 0, stream>>>(srcp, dstp, ls, ld, segmax, segsum, ex, nE, nN);
        scatter_kernel<<<gEPw, BLK, 0, stream>>>(srcp, dstp, ex, segsum, hA, hB, nE, nN);
        finalize_pool_kernel<<<gNH, BLK, 0, stream>>>(hB, bias[layer], batch, pmax, psum, nN);
        accum_z_kernel<<<gZ, BLK, 0, stream>>>(pmax, psum, counts, z);
    }

    // head MLP on z[64, 256]
    fc_kernel<<<(Gd * Hd + BLK - 1) / BLK, BLK, 0, stream>>>(z,  lw1, lb1, t1, 2 * Hd, Hd,      1);
    fc_kernel<<<(Gd * (Hd/2) + BLK - 1) / BLK, BLK, 0, stream>>>(t1, lw2, lb2, t2, Hd,     Hd / 2, 1);
    fc_kernel<<<(Gd * Cd + BLK - 1) / BLK, BLK, 0, stream>>>(t2, lw3, lb3, t3, Hd / 2, Cd,     0);
    logsoftmax_kernel<<<1, Gd, 0, stream>>>(t3, (float*)d_out);
}